// FastRCNNOutputLayers_73804718015062
// MI455X (gfx1250) — compile-verified
//
#include <hip/hip_runtime.h>
#include <stdint.h>

typedef __attribute__((ext_vector_type(2))) float v2f;
typedef __attribute__((ext_vector_type(8))) float v8f;

#define BATCH   8
#define RROWS   1000
#define KCLS    80
#define DDIM    1024
#define NCLS    81        // K+1 (with background)
#define NBOX    320       // K*4
#define NREAL   401       // 81 + 320
#define NPAD    416       // 26 tiles of 16
#define ROWS    (BATCH*RROWS)     // 8000 = 500 M-tiles
#define MTILES  (ROWS/16)         // 500
#define NTILES  (NPAD/16)         // 26
#define MCAND   4096
#define NSORT   131072            // 2^17 >= R*K = 80000
#define TOPK_OUT 100
#define W_IMG   1333.0f
#define H_IMG   800.0f
#define CLS_OFF 1334.0f           // max(W,H)+1
#define SCORE_T 0.01f
#define NMS_T   0.5f
#define SCALE_CLAMP 4.135166556742356f  // ln(1000/16)

// Async global->LDS (gfx1250, ASYNCcnt-tracked) if this toolchain declares it.
#if defined(__has_builtin)
#if __has_builtin(__builtin_amdgcn_global_load_async_to_lds_b128) && \
    __has_builtin(__builtin_amdgcn_s_wait_asynccnt)
#define HAVE_ASYNC_LDS 1
#endif
#endif
#ifndef HAVE_ASYNC_LDS
#define HAVE_ASYNC_LDS 0
#endif

#if HAVE_ASYNC_LDS
#define GLOBAL_AS __attribute__((address_space(1)))
#define LDS_AS    __attribute__((address_space(3)))
typedef int v4i_vec __attribute__((vector_size(4 * sizeof(int))));
#endif

// ---------------------------------------------------------------------------
// 1. Pack W_cls | W_box into one zero-padded [DDIM, NPAD] matrix + bias.
// ---------------------------------------------------------------------------
__global__ void pack_weights(const float* __restrict__ Wc, const float* __restrict__ bc,
                             const float* __restrict__ Wb, const float* __restrict__ bb,
                             float* __restrict__ Wp, float* __restrict__ bp) {
    int t = blockIdx.x * blockDim.x + threadIdx.x;
    if (t < DDIM * NPAD) {
        int k = t / NPAD, n = t % NPAD;
        float v = 0.0f;
        if (n < NCLS)       v = Wc[k * NCLS + n];
        else if (n < NREAL) v = Wb[k * NBOX + (n - NCLS)];
        Wp[t] = v;
    }
    if (t < NPAD) {
        float v = 0.0f;
        if (t < NCLS)       v = bc[t];
        else if (t < NREAL) v = bb[t - NCLS];
        bp[t] = v;
    }
}

// ---------------------------------------------------------------------------
// 2. Fused GEMM: logits[8000,416] = x[8000,1024] @ Wp[1024,416] + bp
//    One 256-thread block (8 waves) per 16-row M-tile. The 16x1024 A panel
//    (contiguous 64KB of x) is staged into LDS once via async global->LDS,
//    then the 8 waves sweep the 26 N-tiles with V_WMMA_F32_16X16X4_F32,
//    reading A fragments from LDS (ds_load_b64) and B columns from L2.
// ---------------------------------------------------------------------------
__global__ void __launch_bounds__(256)
gemm_wmma_f32(const float* __restrict__ x,
              const float* __restrict__ Wp,
              const float* __restrict__ bp,
              float* __restrict__ out) {
    __shared__ float a_lds[16 * DDIM];           // 64 KB A panel

    const int mTile = blockIdx.x;                // 500 blocks
    const float* src = x + (size_t)mTile * 16 * DDIM;   // contiguous 16 rows

    // Stage A panel: 16384 floats = 4096 float4, 16 per thread.
#pragma unroll 4
    for (int i = threadIdx.x; i < 16 * DDIM / 4; i += 256) {
#if HAVE_ASYNC_LDS
        __builtin_amdgcn_global_load_async_to_lds_b128(
            (GLOBAL_AS v4i_vec*)(src + (size_t)i * 4),
            (LDS_AS v4i_vec*)(a_lds + (size_t)i * 4),
            /*offset=*/0, /*cpol=*/0);
#else
        ((float4*)a_lds)[i] = ((const float4*)src)[i];
#endif
    }
#if HAVE_ASYNC_LDS
    __builtin_amdgcn_s_wait_asynccnt(0);
#endif
    __syncthreads();

    const int wv   = threadIdx.x >> 5;           // 8 waves sweep n-tiles
    const int lane = threadIdx.x & 31;
    const int half = lane >> 4;                  // 0: K-pair {0,1}, 1: {2,3}
    const int lrow = lane & 15;                  // M for A-frag, N for B/C/D

    const float* arow = a_lds + lrow * DDIM + 2 * half;   // LDS reads

    for (int nt = wv; nt < NTILES; nt += 8) {
        const float* bcol = Wp + (size_t)(2 * half) * NPAD + nt * 16 + lrow;
        v8f acc = {};
        for (int k = 0; k < DDIM; k += 4) {
            v2f a, b;
            a.x = arow[k];
            a.y = arow[k + 1];
            b.x = bcol[(size_t)k * NPAD];
            b.y = bcol[(size_t)(k + 1) * NPAD];
            acc = __builtin_amdgcn_wmma_f32_16x16x4_f32(
                /*neg_a=*/false, a, /*neg_b=*/false, b,
                /*c_mod=*/(short)0, acc, /*reuse_a=*/false, /*reuse_b=*/false);
        }
        int ncol = nt * 16 + lrow;
        float bias = bp[ncol];
#pragma unroll
        for (int v = 0; v < 8; ++v) {
            int m = mTile * 16 + v + 8 * half;   // C/D: VGPR v -> M=v (+8 hi half)
            out[(size_t)m * NPAD + ncol] = acc[v] + bias;
        }
    }
}

// ---------------------------------------------------------------------------
// 3. Sort keys: zero-init + softmax/threshold pack.
//    key = ord(score)<<32 | (0xFFFFFFFF - flat_idx)  (descending sort ==
//    top_k with lowest-index tie-break). Invalid -> key 0.
// ---------------------------------------------------------------------------
__global__ void init_keys(unsigned long long* __restrict__ keys) {
    size_t t = (size_t)blockIdx.x * blockDim.x + threadIdx.x;
    if (t < (size_t)BATCH * NSORT) keys[t] = 0ull;
}

__global__ void softmax_keys(const float* __restrict__ logits,
                             unsigned long long* __restrict__ keys) {
    int r = blockIdx.x * blockDim.x + threadIdx.x;
    if (r >= ROWS) return;
    const float* row = logits + (size_t)r * NPAD;
    float mx = row[0];
    for (int c = 1; c < NCLS; ++c) mx = fmaxf(mx, row[c]);
    float sum = 0.0f;
    for (int c = 0; c < NCLS; ++c) sum += __expf(row[c] - mx);
    float inv = 1.0f / sum;
    int img = r / RROWS, rl = r % RROWS;
    unsigned long long* kb = keys + (size_t)img * NSORT + (size_t)rl * KCLS;
    for (int c = 0; c < KCLS; ++c) {             // background (c==80) dropped
        float p = __expf(row[c] - mx) * inv;
        unsigned long long key = 0ull;
        if (p > SCORE_T) {
            unsigned int ord = __float_as_uint(p);          // monotonic for p>0
            unsigned int idx = (unsigned)(rl * KCLS + c);
            key = ((unsigned long long)ord << 32) |
                  (unsigned long long)(0xFFFFFFFFu - idx);
        }
        kb[c] = key;
    }
}

// ---------------------------------------------------------------------------
// 4. Global bitonic sort (descending), per image over NSORT slots.
// ---------------------------------------------------------------------------
__global__ void bitonic_pass(unsigned long long* __restrict__ keys, int k, int j) {
    size_t t = (size_t)blockIdx.x * blockDim.x + threadIdx.x;
    if (t >= (size_t)BATCH * NSORT) return;
    int img = (int)(t / NSORT);
    int i = (int)(t % NSORT);
    int l = i ^ j;
    if (l <= i) return;
    unsigned long long* kk = keys + (size_t)img * NSORT;
    unsigned long long a = kk[i], b = kk[l];
    bool desc = ((i & k) == 0);
    if (desc ? (a < b) : (a > b)) { kk[i] = b; kk[l] = a; }
}

// ---------------------------------------------------------------------------
// 5. Decode top-MCAND candidates: Box2Box transform + clip.
// ---------------------------------------------------------------------------
__global__ void decode_cands(const unsigned long long* __restrict__ keys,
                             const float* __restrict__ logits,
                             const float* __restrict__ props,
                             float* __restrict__ cboxes,   // [B][M][4] (no class offset)
                             float* __restrict__ cscore,   // [B][M]
                             int* __restrict__ ccls,       // [B][M]
                             int* __restrict__ cvalid) {   // [B][M]
    int t = blockIdx.x * blockDim.x + threadIdx.x;
    if (t >= BATCH * MCAND) return;
    int img = t / MCAND, slot = t % MCAND;
    unsigned long long key = keys[(size_t)img * NSORT + slot];
    float score = -1.0f; int cls = 0; int valid = 0;
    float b0 = 0.f, b1 = 0.f, b2 = 0.f, b3 = 0.f;
    if (key != 0ull) {
        unsigned int ord = (unsigned)(key >> 32);
        unsigned int idx = 0xFFFFFFFFu - (unsigned)(key & 0xFFFFFFFFull);
        score = __uint_as_float(ord);
        int r = (int)(idx / KCLS);
        cls = (int)(idx % KCLS);
        const float* d  = logits + (size_t)(img * RROWS + r) * NPAD + NCLS + cls * 4;
        const float* pb = props  + (size_t)(img * RROWS + r) * 4;
        float w  = pb[2] - pb[0], h = pb[3] - pb[1];
        float cx = pb[0] + 0.5f * w, cy = pb[1] + 0.5f * h;
        float dx = d[0] * 0.1f, dy = d[1] * 0.1f;
        float dw = fminf(d[2] * 0.2f, SCALE_CLAMP);
        float dh = fminf(d[3] * 0.2f, SCALE_CLAMP);
        float pcx = dx * w + cx, pcy = dy * h + cy;
        float pw = __expf(dw) * w, ph = __expf(dh) * h;
        b0 = fminf(fmaxf(pcx - 0.5f * pw, 0.0f), W_IMG);
        b1 = fminf(fmaxf(pcy - 0.5f * ph, 0.0f), H_IMG);
        b2 = fminf(fmaxf(pcx + 0.5f * pw, 0.0f), W_IMG);
        b3 = fminf(fmaxf(pcy + 0.5f * ph, 0.0f), H_IMG);
        valid = 1;
    }
    float* cb = cboxes + (size_t)t * 4;
    cb[0] = b0; cb[1] = b1; cb[2] = b2; cb[3] = b3;
    cscore[t] = score;
    ccls[t]   = cls;
    cvalid[t] = valid;
}

// ---------------------------------------------------------------------------
// 6. Greedy per-class NMS (class offset trick), one workgroup per image.
// ---------------------------------------------------------------------------
__global__ void nms_kernel(const float* __restrict__ cboxes,
                           const int* __restrict__ ccls,
                           const int* __restrict__ cvalid,
                           int* __restrict__ keep) {
    int img = blockIdx.x;
    __shared__ unsigned char sup[MCAND];
    const float* B = cboxes + (size_t)img * MCAND * 4;
    const int* C = ccls  + (size_t)img * MCAND;
    const int* V = cvalid + (size_t)img * MCAND;
    int* KP = keep + (size_t)img * MCAND;
    for (int s = threadIdx.x; s < MCAND; s += blockDim.x) sup[s] = 0;
    __syncthreads();
    for (int i = 0; i < MCAND; ++i) {
        bool keep_i = (V[i] != 0) && (sup[i] == 0);
        if (threadIdx.x == 0) KP[i] = keep_i ? 1 : 0;
        if (keep_i) {
            float off_i = (float)C[i] * CLS_OFF;
            float x1 = B[i * 4 + 0] + off_i, y1 = B[i * 4 + 1] + off_i;
            float x2 = B[i * 4 + 2] + off_i, y2 = B[i * 4 + 3] + off_i;
            float ai = (x2 - x1) * (y2 - y1);
            for (int j = i + 1 + (int)threadIdx.x; j < MCAND; j += (int)blockDim.x) {
                if (sup[j]) continue;
                float off_j = (float)C[j] * CLS_OFF;
                float u1 = B[j * 4 + 0] + off_j, v1 = B[j * 4 + 1] + off_j;
                float u2 = B[j * 4 + 2] + off_j, v2 = B[j * 4 + 3] + off_j;
                float aj = (u2 - u1) * (v2 - v1);
                float iw = fminf(x2, u2) - fmaxf(x1, u1);
                float ih = fminf(y2, v2) - fmaxf(y1, v1);
                iw = fmaxf(iw, 0.0f); ih = fmaxf(ih, 0.0f);
                float inter = iw * ih;
                float iou = inter / (ai + aj - inter + 1e-12f);
                if (iou > NMS_T) sup[j] = 1;
            }
        }
        __syncthreads();
    }
}

// ---------------------------------------------------------------------------
// 7. Finalize: candidates are sorted by score, so first 100 kept == top-100.
//    d_out layout: boxes[8,100,4] | scores[8,100] | classes[8,100] | valid[8,100]
// ---------------------------------------------------------------------------
__global__ void finalize(const float* __restrict__ cboxes,
                         const float* __restrict__ cscore,
                         const int* __restrict__ ccls,
                         const int* __restrict__ keep,
                         float* __restrict__ out) {
    int img = blockIdx.x;
    if (threadIdx.x != 0) return;
    float* ob = out;
    float* os = out + BATCH * TOPK_OUT * 4;
    float* oc = os + BATCH * TOPK_OUT;
    float* ov = oc + BATCH * TOPK_OUT;
    int cnt = 0;
    for (int s = 0; s < MCAND && cnt < TOPK_OUT; ++s) {
        if (keep[img * MCAND + s]) {
            int o = img * TOPK_OUT + cnt;
            for (int q = 0; q < 4; ++q) ob[o * 4 + q] = cboxes[((size_t)img * MCAND + s) * 4 + q];
            os[o] = cscore[img * MCAND + s];
            oc[o] = (float)ccls[img * MCAND + s];
            ov[o] = 1.0f;
            ++cnt;
        }
    }
    for (; cnt < TOPK_OUT; ++cnt) {
        int o = img * TOPK_OUT + cnt;
        for (int q = 0; q < 4; ++q) ob[o * 4 + q] = 0.0f;
        os[o] = -1.0f;
        oc[o] = 0.0f;
        ov[o] = 0.0f;
    }
}

// ---------------------------------------------------------------------------
extern "C" void kernel_launch(void* const* d_in, const int* in_sizes, int n_in,
                              void* d_out, int out_size, void* d_ws, size_t ws_size,
                              hipStream_t stream) {
    const float* x   = (const float*)d_in[0];   // [8000,1024]
    const float* Wc  = (const float*)d_in[1];   // [1024,81]
    const float* bc  = (const float*)d_in[2];   // [81]
    const float* Wb  = (const float*)d_in[3];   // [1024,320]
    const float* bb  = (const float*)d_in[4];   // [320]
    const float* pr  = (const float*)d_in[5];   // [8000,4]
    float* out = (float*)d_out;
    (void)in_sizes; (void)n_in; (void)out_size; (void)ws_size;

    // Workspace carve-up (256B aligned)
    char* w = (char*)d_ws;
    size_t off = 0;
    auto carve = [&](size_t bytes) -> char* {
        char* p = w + off;
        off = (off + bytes + 255) & ~(size_t)255;
        return p;
    };
    float* Wp                 = (float*)carve((size_t)DDIM * NPAD * sizeof(float));
    float* bp                 = (float*)carve((size_t)NPAD * sizeof(float));
    float* logits             = (float*)carve((size_t)ROWS * NPAD * sizeof(float));
    unsigned long long* keys  = (unsigned long long*)carve((size_t)BATCH * NSORT * 8);
    float* cboxes             = (float*)carve((size_t)BATCH * MCAND * 4 * sizeof(float));
    float* cscore             = (float*)carve((size_t)BATCH * MCAND * sizeof(float));
    int*   ccls               = (int*)carve((size_t)BATCH * MCAND * sizeof(int));
    int*   cvalid             = (int*)carve((size_t)BATCH * MCAND * sizeof(int));
    int*   keep               = (int*)carve((size_t)BATCH * MCAND * sizeof(int));

    // 1. Pack weights
    {
        int n = DDIM * NPAD;
        pack_weights<<<(n + 255) / 256, 256, 0, stream>>>(Wc, bc, Wb, bb, Wp, bp);
    }
    // 2. WMMA GEMM: one block per M-tile, 8 waves sweep the 26 N-tiles
    gemm_wmma_f32<<<MTILES, 256, 0, stream>>>(x, Wp, bp, logits);
    // 3. Keys
    {
        size_t n = (size_t)BATCH * NSORT;
        init_keys<<<(unsigned)((n + 255) / 256), 256, 0, stream>>>(keys);
        softmax_keys<<<(ROWS + 127) / 128, 128, 0, stream>>>(logits, keys);
    }
    // 4. Bitonic sort (descending), 153 passes
    {
        unsigned blocks = (unsigned)(((size_t)BATCH * NSORT + 255) / 256);
        for (int k = 2; k <= NSORT; k <<= 1)
            for (int j = k >> 1; j > 0; j >>= 1)
                bitonic_pass<<<blocks, 256, 0, stream>>>(keys, k, j);
    }
    // 5. Decode candidates
    {
        int n = BATCH * MCAND;
        decode_cands<<<(n + 255) / 256, 256, 0, stream>>>(keys, logits, pr,
                                                          cboxes, cscore, ccls, cvalid);
    }
    // 6. NMS
    nms_kernel<<<BATCH, 1024, 0, stream>>>(cboxes, ccls, cvalid, keep);
    // 7. Finalize
    finalize<<<BATCH, 32, 0, stream>>>(cboxes, cscore, ccls, keep, out);
}